// Label_Smooth_NLLLoss_41824391528713
// MI455X (gfx1250) — compile-verified
//
#include <hip/hip_runtime.h>
#include <math.h>

#define VOCAB        32000
#define NTOK         4096
#define PADIDX       0
#define LSM          0.1f
#define CONF         0.9f

#define WAVES        8          // 256 threads / wave32
#define CHUNK_FLOATS 128        // 32 lanes * 4 floats
#define CHUNK_BYTES  512
#define NCHUNK       (VOCAB / CHUNK_FLOATS)   // 250
#define DEPTH        4          // async pipeline depth per wave (2 KB in flight)

// Generic pointer to a __shared__ object = {LDS aperture hi bits, 32-bit LDS offset};
// low 32 bits are exactly the workgroup-relative LDS byte address the async ops want.
__device__ __forceinline__ unsigned lds_byte_addr(const void* p) {
  return (unsigned)(unsigned long long)p;
}

// Issue one 512B chunk (16B/lane) into LDS, non-temporal (read-once stream > L2).
__device__ __forceinline__ void issue_async_chunk(unsigned ldsAddr, unsigned goff,
                                                  unsigned long long base) {
  asm volatile("s_wait_dscnt 0x0\n\t"       // prior consume's ds reads drained
               "global_load_async_to_lds_b128 %0, %1, %2 th:TH_LOAD_NT"
               :: "v"(ldsAddr), "v"(goff), "s"(base) : "memory");
}

// ---------------------------------------------------------------------------
// Kernel 1: per-row loss via depth-4 async-to-LDS streaming reduction
// ---------------------------------------------------------------------------
__global__ void __launch_bounds__(256)
row_loss_kernel(const float* __restrict__ out, const int* __restrict__ tgt,
                float* __restrict__ rowloss) {
  __shared__ float lds[WAVES * DEPTH * CHUNK_FLOATS];   // 16 KB
  __shared__ float waveSum[WAVES];

  const int row  = blockIdx.x;
  const int tid  = threadIdx.x;
  const int wave = tid >> 5;
  const int lane = tid & 31;

  const unsigned long long rowBase =
      (unsigned long long)(out + (size_t)row * VOCAB);   // SGPR base (GVS mode)

  // Per-wave buffers are contiguous: buffer b at waveLds + b*512 bytes.
  const unsigned waveLds =
      lds_byte_addr(&lds[wave * DEPTH * CHUNK_FLOATS + lane * 4]);

  // chunks for this wave: c_i = wave + i*WAVES, i = 0..k-1
  const int k = (NCHUNK - wave + WAVES - 1) / WAVES;     // 31 or 32 (k >= DEPTH)

  // --- prologue: fill all DEPTH buffers ---
#pragma unroll
  for (int i = 0; i < DEPTH; ++i) {
    const unsigned goff =
        (unsigned)(wave + i * WAVES) * CHUNK_BYTES + (unsigned)lane * 16u;
    issue_async_chunk(waveLds + (unsigned)i * CHUNK_BYTES, goff, rowBase);
  }

  float ax = 0.f, ay = 0.f, az = 0.f, aw = 0.f;
  int i = 0;
  // --- steady state: wait oldest -> consume -> refill the freed buffer ---
  for (; i + DEPTH < k; ++i) {
    asm volatile("s_wait_asynccnt 0x3" ::: "memory");    // chunk i landed
    const int b = i & (DEPTH - 1);
    const float4 v = *reinterpret_cast<const float4*>(
        &lds[(wave * DEPTH + b) * CHUNK_FLOATS + lane * 4]);   // ds_load_b128
    ax += v.x; ay += v.y; az += v.z; aw += v.w;
    const unsigned goff =
        (unsigned)(wave + (i + DEPTH) * WAVES) * CHUNK_BYTES + (unsigned)lane * 16u;
    issue_async_chunk(waveLds + (unsigned)b * CHUNK_BYTES, goff, rowBase);
  }
  // --- tail: drain and consume the last DEPTH chunks ---
  asm volatile("s_wait_asynccnt 0x0" ::: "memory");
  for (; i < k; ++i) {
    const int b = i & (DEPTH - 1);
    const float4 v = *reinterpret_cast<const float4*>(
        &lds[(wave * DEPTH + b) * CHUNK_FLOATS + lane * 4]);
    ax += v.x; ay += v.y; az += v.z; aw += v.w;
  }

  // wave32 shuffle reduction of the per-lane partials
  float s = (ax + ay) + (az + aw);
#pragma unroll
  for (int o = 16; o > 0; o >>= 1) s += __shfl_down(s, o, 32);
  if (lane == 0) waveSum[wave] = s;
  __syncthreads();

  if (tid == 0) {
    float rowSum = 0.f;
#pragma unroll
    for (int w = 0; w < WAVES; ++w) rowSum += waveSum[w];
    const int t = tgt[row];
    float loss = 0.f;
    if (t != PADIDX) {
      const float eps = LSM / (float)(VOCAB - 2);
      const float C   = LSM * logf(eps) + CONF * logf(CONF);  // (V-2)*eps = LSM
      const float o0  = out[(size_t)row * VOCAB + PADIDX];
      const float ot  = out[(size_t)row * VOCAB + t];
      loss = C + eps * o0 + (eps - CONF) * ot - eps * rowSum;
    }
    rowloss[row] = loss;
  }
}

// ---------------------------------------------------------------------------
// Kernel 2: single-wave final reduction, finishing through v_wmma_f32_16x16x4_f32
// A holds the 32 lane partials in K-slots 0 (lanes 0-15) and 2 (lanes 16-31);
// B = ones  =>  D[m][n] = p_m + p_{m+16} for every column n.
// ---------------------------------------------------------------------------
typedef __attribute__((ext_vector_type(2))) float v2f;
typedef __attribute__((ext_vector_type(8))) float v8f;

__global__ void __launch_bounds__(32)
final_reduce_kernel(const float* __restrict__ rowloss, float* __restrict__ outv) {
  const int lane = threadIdx.x;
  float ax = 0.f, ay = 0.f, az = 0.f, aw = 0.f;
  const float4* rl4 = reinterpret_cast<const float4*>(rowloss);
#pragma unroll 4
  for (int i = lane; i < NTOK / 4; i += 32) {   // uniform trip count: EXEC stays all-1s
    const float4 v = rl4[i];
    ax += v.x; ay += v.y; az += v.z; aw += v.w;
  }
  const float p = (ax + ay) + (az + aw);

  v2f a; a.x = p;   a.y = 0.f;   // A[m][0]=p_m (lanes 0-15), A[m][2]=p_{m+16} (lanes 16-31)
  v2f b; b.x = 1.f; b.y = 1.f;   // B = all ones
  v8f cacc = {};
  v8f d = __builtin_amdgcn_wmma_f32_16x16x4_f32(
      /*neg_a=*/false, a, /*neg_b=*/false, b,
      /*c_mod=*/(short)0, cacc, /*reuse_a=*/false, /*reuse_b=*/false);

  // Column 0 of D: lanes 0..15 hold rows m=0..7 in their 8 VGPRs, lanes 16..31 rows 8..15.
  float s = ((d[0] + d[1]) + (d[2] + d[3])) + ((d[4] + d[5]) + (d[6] + d[7]));
  s += __shfl_xor(s, 16, 32);    // combine the two half-column sums
  if (lane == 0) outv[0] = s;
}

// ---------------------------------------------------------------------------
extern "C" void kernel_launch(void* const* d_in, const int* in_sizes, int n_in,
                              void* d_out, int out_size, void* d_ws, size_t ws_size,
                              hipStream_t stream) {
  const float* out = (const float*)d_in[0];
  const int*   tgt = (const int*)d_in[1];
  float* ws = (float*)d_ws;                      // 4096 floats of scratch

  row_loss_kernel<<<NTOK, 256, 0, stream>>>(out, tgt, ws);
  final_reduce_kernel<<<1, 32, 0, stream>>>(ws, (float*)d_out);

  (void)in_sizes; (void)n_in; (void)out_size; (void)ws_size;
}